// Atmoshpere_69363721830844
// MI455X (gfx1250) — compile-verified
//
#include <hip/hip_runtime.h>
#include <math.h>
#include <stdint.h>

#define NZc 32
#define NYc 360
#define NXc 720
#define SU (NYc*(NXc+1))   // per-k stride of u-shaped arrays
#define SV ((NYc+1)*NXc)   // per-k stride of v-shaped arrays
#define ST (NYc*NXc)       // per-k stride of theta-shaped arrays
#define TB 256             // tile width / block size
#define LW 260             // padded LDS row width

static constexpr double dPI = 3.14159265358979323846;
static constexpr double dRE = 6400000.0;
static constexpr double dDX = 2.0*dPI/(double)NXc;
static constexpr double dDY = dPI/(double)NYc;
static constexpr double dDA = dRE*dDX*(dRE*dDY);

static constexpr float fDX  = (float)dDX;
static constexpr float fDY  = (float)dDY;
static constexpr float fDA  = (float)dDA;
#define fRE   6400000.0f
#define fCPD  1004.67f
#define fDIFF 1.0e9f
#define K48   (1.0f/48.0f)
#define K96   (1.0f/96.0f)

__device__ __forceinline__ int wx(int t) { return (t + NXc) % NXc; }  // wrap width 720
__device__ __forceinline__ int wy(int t) { return (t + NYc) % NYc; }  // wrap height 360

// ---- gfx1250 async global->LDS copy (1 dword per lane, tracked by ASYNCcnt)
__device__ __forceinline__ void g2l(const float* g, float* l) {
    asm volatile("global_load_async_to_lds_b32 %0, %1, off"
                 :: "v"((uint32_t)(uintptr_t)l),
                    "v"((uint64_t)(uintptr_t)g)
                 : "memory");
}
__device__ __forceinline__ void async_fence() {
    asm volatile("s_wait_asynccnt 0" ::: "memory");
    __syncthreads();
}

// ---------------------------------------------------------------- cos(Y)
__global__ void cosy_kernel(const float* __restrict__ Y, float* __restrict__ cosY) {
    int i = blockIdx.x * blockDim.x + threadIdx.x;
    if (i <= NYc) cosY[i] = cosf(Y[i]);
}

// ------------------------------------------------- 2D flux coefficients
__global__ void coeff_kernel(const float* __restrict__ pi2d,
                             const float* __restrict__ cosY,
                             float* __restrict__ aF, float* __restrict__ bG) {
    int idx = blockIdx.x * blockDim.x + threadIdx.x;
    if (idx < NYc * (NXc + 1)) {                  // aF: (NY, NX+1)
        int j = idx / (NXc + 1), i = idx - j * (NXc + 1);
        aF[idx] = 0.25f * fRE * fDY * (pi2d[j*NXc + wx(i-1)] + pi2d[j*NXc + wx(i)]);
    }
    if (idx < (NYc + 1) * NXc) {                  // bG: (NY+1, NX)
        int j = idx / NXc, i = idx - j * NXc;
        bG[idx] = 0.25f * fRE * fDX * cosY[j] *
                  (pi2d[wy(j-1)*NXc + i] + pi2d[wy(j)*NXc + i]);
    }
}

// ---------------------------------------------------------------- pi1
__global__ void pi_kernel(const float* __restrict__ pi0,
                          const float* __restrict__ u, const float* __restrict__ v,
                          const float* __restrict__ aF, const float* __restrict__ bG,
                          const float* __restrict__ dtp, float dtmul,
                          float* __restrict__ out) {
    int idx = blockIdx.x * blockDim.x + threadIdx.x;
    if (idx >= ST) return;
    int j = idx / NXc, i = idx - j * NXc;
    float ai  = aF[j*(NXc+1) + i], aip = aF[j*(NXc+1) + i + 1];
    float bj  = bG[j*NXc + i],     bjp = bG[(j+1)*NXc + i];
    const float* up = u + j*(NXc+1) + i;
    const float* vp = v + j*NXc + i;
    float sum = 0.0f;
    #pragma unroll 4
    for (int k = 0; k < NZc; ++k) {
        __builtin_prefetch(up + 2*SU, 0, 0);
        __builtin_prefetch(vp + 2*SV, 0, 0);
        sum += aip*up[1] - ai*up[0] + bjp*vp[NXc] - bj*vp[0];
        up += SU; vp += SV;
    }
    float dt = dtp[0] * dtmul;
    out[idx] = pi0[idx] - dt / fDA * (sum * (1.0f/32.0f));
}

// ======================= theta1 (x-tiled, async-LDS staged) ============
__device__ __forceinline__ void stage_theta(const float* th, const float* u, int k,
        int jm, int j, int jp, int x0, int t, float* sT, float* sUu) {
    const int oT = k*ST, oU = k*SU;
    #pragma unroll
    for (int r = 0; r < 3; ++r) {                  // th rows jm,j,jp; 720-wrap
        const int row = (r == 0) ? jm : ((r == 1) ? j : jp);
        const float* src = th + oT + row*NXc;
        float* dst = sT + r*LW;
        for (int c = t; c < 258; c += TB) {
            int col = x0 - 1 + c;
            if (col < 0) col += NXc; else if (col >= NXc) col -= NXc;
            g2l(src + col, dst + c);
        }
    }
    const float* usrc = u + oU + j*(NXc+1);        // u row j, cols x0..x0+256 (clamped)
    for (int c = t; c < 257; c += TB) {
        int col = x0 + c; if (col > NXc) col = NXc;
        g2l(usrc + col, sUu + c);
    }
}

__global__ void theta_kernel(const float* __restrict__ th,  const float* __restrict__ th0,
                             const float* __restrict__ pi2d, const float* __restrict__ pi0,
                             const float* __restrict__ pi1v,
                             const float* __restrict__ u, const float* __restrict__ v,
                             const float* __restrict__ w,
                             const float* __restrict__ aF, const float* __restrict__ bG,
                             const float* __restrict__ dtp, float dtmul,
                             float* __restrict__ out) {
    __shared__ float shT[2*3*LW];
    __shared__ float shU[2*LW];
    const int t  = threadIdx.x;
    const int x0 = blockIdx.x * TB;
    const int j  = blockIdx.y;
    const int i  = x0 + t;
    const bool act = (i < NXc);
    const int ii  = act ? i : (NXc - 1);
    const int jm = wy(j-1), jp = wy(j+1);
    const int idx = j*NXc + ii;
    float ai  = aF[j*(NXc+1) + ii], aip = aF[j*(NXc+1) + ii + 1];
    float bj  = bG[j*NXc + ii],     bjp = bG[(j+1)*NXc + ii];
    float piA   = pi2d[idx] * fDA;
    float ratio = pi0[idx] / pi1v[idx];
    float dt    = dtp[0] * dtmul;
    float coef  = dt / fDA / pi1v[idx];
    float tc_prev = 0.0f;
    float tc      = th[idx];
    stage_theta(th, u, 0, jm, j, jp, x0, t, shT, shU);
    async_fence();
    int buf = 0;
    for (int k = 0; k < NZc; ++k) {
        if (k + 1 < NZc)
            stage_theta(th, u, k+1, jm, j, jp, x0, t,
                        shT + (buf^1)*3*LW, shU + (buf^1)*LW);
        if (act) {
            const float* T0 = shT + buf*3*LW;      // row jm
            const float* T1 = T0 + LW;             // row j
            const float* T2 = T1 + LW;             // row jp
            const float* Uj = shU + buf*LW;
            const int oT = k*ST, oV = k*SV;
            float tc_next = (k < NZc-1) ? th[oT + ST + idx] : 0.0f;
            float tim = T1[t],   tip_ = T1[t+2];
            float tjm = T0[t+1], tjp  = T2[t+1];
            float uL = Uj[t], uR = Uj[t+1];
            float vB = v[oV + j*NXc + ii], vT = v[oV + (j+1)*NXc + ii];
            float adv = 0.5f * ((aip*uR*0.5f*(tc + tip_) - ai*uL*0.5f*(tim + tc))
                              + (bjp*vT*0.5f*(tc + tjp)  - bj*vB*0.5f*(tjm + tc)));
            float t0  = th0[oT + idx];
            float t2  = piA * w[oT + idx] * t0 * 32.0f;
            float lap = tc_prev + tc_next + tjm + tjp + tim + tip_ - 6.0f*tc;
            out[oT + idx] = ratio * t0 + coef * (adv + t2 + fDIFF * lap);
            tc_prev = tc; tc = tc_next;
        }
        async_fence();
        buf ^= 1;
    }
}

// ======================= u1 (x-tiled, async-LDS staged) ================
__device__ __forceinline__ void stage_u_tiles(const float* u, const float* v, int k,
        int jm, int j, int jp, int gm, int g2, int x0, int t,
        float* sUb, float* sVb) {
    const int oU = k*SU, oV = k*SV;
    #pragma unroll
    for (int r = 0; r < 3; ++r) {                  // u rows jm,j,jp; 721-wrap
        const int row = (r == 0) ? jm : ((r == 1) ? j : jp);
        const float* src = u + oU + row*(NXc+1);
        float* dst = sUb + r*LW;
        for (int c = t; c < 259; c += TB) {
            int col = x0 - 1 + c;
            if (col < 0) col += NXc+1; else if (col > NXc) col -= NXc+1;
            g2l(src + col, dst + c);
        }
    }
    #pragma unroll
    for (int r = 0; r < 4; ++r) {                  // v rows gm,j,j+1,g2; 720-wrap
        const int row = (r == 0) ? gm : ((r == 1) ? j : ((r == 2) ? j+1 : g2));
        const float* src = v + oV + row*NXc;
        float* dst = sVb + r*LW;
        for (int c = t; c < 257; c += TB) {
            int col = x0 - 1 + c;
            if (col < 0) col += NXc; else if (col >= NXc) col -= NXc;
            g2l(src + col, dst + c);
        }
    }
}

__global__ void u_kernel(const float* __restrict__ u,  const float* __restrict__ u0,
                         const float* __restrict__ v,
                         const float* __restrict__ pi2d, const float* __restrict__ pi0,
                         const float* __restrict__ pi1v,
                         const float* __restrict__ w,   const float* __restrict__ phi,
                         const float* __restrict__ th,
                         const float* __restrict__ aF,  const float* __restrict__ bG,
                         const float* __restrict__ sg,  const float* __restrict__ Pu,
                         const float* __restrict__ Pp,
                         const float* __restrict__ dtp, float dtmul,
                         float* __restrict__ out) {
    __shared__ float shU[2*3*LW];
    __shared__ float shV[2*4*LW];
    const int t  = threadIdx.x;
    const int x0 = blockIdx.x * TB;
    const int j  = blockIdx.y;
    const int i  = x0 + t;
    const bool act = (i <= NXc);                   // width NX+1
    const int ii = act ? i : NXc;
    const int jm = wy(j-1), jp = wy(j+1);
    const int gm = (j == 0)       ? NYc : j - 1;   // G-row wrap (361)
    const int g2 = (j == NYc - 1) ? 0   : j + 2;
    const int c0 = wx(ii-1), c1 = wx(ii);
    const int c0p = c0 + 1, c1p = c1 + 1;
    // staged-slot indices (u staged with width-721 wrap; seams where 720/721 diverge)
    int sc0 = t, sc0p = t + 1, sc1 = t + 1, sc1p = t + 2;
    const bool seam0 = (ii == 0);
    if (seam0)     { sc0p = 0; }                   // col 720 lives at slot 0
    if (ii == NXc) { sc1 = t + 2; sc1p = t + 3; }
    // hoisted coefficients
    const int am = jm*(NXc+1), aj = j*(NXc+1), ap = jp*(NXc+1);
    float a_m0 = aF[am+c0], a_m0p = aF[am+c0p], a_m1 = aF[am+c1], a_m1p = aF[am+c1p];
    float a_j0 = aF[aj+c0], a_j0p = aF[aj+c0p], a_j1 = aF[aj+c1], a_j1p = aF[aj+c1p];
    float a_p0 = aF[ap+c0], a_p0p = aF[ap+c0p], a_p1 = aF[ap+c1], a_p1p = aF[ap+c1p];
    float b_0a = bG[gm*NXc+c0],    b_0b = bG[gm*NXc+c1];
    float b_1a = bG[j*NXc+c0],     b_1b = bG[j*NXc+c1];
    float b_2a = bG[(j+1)*NXc+c0], b_2b = bG[(j+1)*NXc+c1];
    float b_3a = bG[g2*NXc+c0],    b_3b = bG[g2*NXc+c1];
    float pim0 = pi2d[jm*NXc+c0], pim1 = pi2d[jm*NXc+c1];
    float pip0 = pi2d[jp*NXc+c0], pip1 = pi2d[jp*NXc+c1];
    float dpix = pi2d[j*NXc+c1] - pi2d[j*NXc+c0];
    float p0dA = (fDA*0.03125f)*(pi0[jm*NXc+c0]+pi0[jm*NXc+c1]+pi0[jp*NXc+c0]+pi0[jp*NXc+c1]);
    float p1dA = (fDA*0.03125f)*(pi1v[jm*NXc+c0]+pi1v[jm*NXc+c1]+pi1v[jp*NXc+c0]+pi1v[jp*NXc+c1]);
    float dt = dtp[0] * dtmul;
    float u_zm = 0.0f;
    float u_c  = u[j*(NXc+1) + ii];
    stage_u_tiles(u, v, 0, jm, j, jp, gm, g2, x0, t, shU, shV);
    async_fence();
    int buf = 0;
    for (int k = 0; k < NZc; ++k) {
        if (k + 1 < NZc)
            stage_u_tiles(u, v, k+1, jm, j, jp, gm, g2, x0, t,
                          shU + (buf^1)*3*LW, shV + (buf^1)*4*LW);
        if (act) {
            const float* U0 = shU + buf*3*LW;      // row jm
            const float* U1 = U0 + LW;             // row j
            const float* U2 = U1 + LW;             // row jp
            const float* V0 = shV + buf*4*LW;      // row gm
            const float* V1 = V0 + LW;             // row j
            const float* V2 = V1 + LW;             // row j+1
            const float* V3 = V2 + LW;             // row g2
            const int oU = k*SU, oT = k*ST;
            const int rjg = oU + j*(NXc+1);
            float u_zp = (k < NZc-1) ? u[rjg + SU + ii] : 0.0f;
            float um0, uj0, up0;
            if (seam0) {                           // col 719 not staged for lane 0
                um0 = u[oU + jm*(NXc+1) + c0];
                uj0 = u[rjg + c0];
                up0 = u[oU + jp*(NXc+1) + c0];
            } else { um0 = U0[sc0]; uj0 = U1[sc0]; up0 = U2[sc0]; }
            float um0p = U0[sc0p], uj0p = U1[sc0p], up0p = U2[sc0p];
            float um1  = U0[sc1],  uj1  = U1[sc1],  up1  = U2[sc1];
            float um1p = U0[sc1p], uj1p = U1[sc1p], up1p = U2[sc1p];
            float um_i = U0[t+1], up_i = U2[t+1];
            float u_iL = U1[t],   u_iR = U1[t+2];
            float Fm0=a_m0*um0, Fm0p=a_m0p*um0p, Fm1=a_m1*um1, Fm1p=a_m1p*um1p;
            float Fj0=a_j0*uj0, Fj0p=a_j0p*uj0p, Fj1=a_j1*uj1, Fj1p=a_j1p*uj1p;
            float Fp0=a_p0*up0, Fp0p=a_p0p*up0p, Fp1=a_p1*up1, Fp1p=a_p1p*up1p;
            float G0a=b_0a*V0[t], G0b=b_0b*V0[t+1];
            float G1a=b_1a*V1[t], G1b=b_1b*V1[t+1];
            float G2a=b_2a*V2[t], G2b=b_2b*V2[t+1];
            float G3a=b_3a*V3[t], G3b=b_3b*V3[t+1];
            float Bu0 = (K48*(Fm0+Fm0p+Fp0+Fp0p)) * (0.5f*(uj0+uj0p));
            float Bu1 = (K48*(Fm1+Fm1p+Fp1+Fp1p)) * (0.5f*(uj1+uj1p));
            float Cu_j  = (K48*(G0a+G0b+G2a+G2b)) * (0.5f*(um_i + u_c));
            float Cu_jp = (K48*(G1a+G1b+G3a+G3b)) * (0.5f*(u_c + up_i));
            float Dj0 = 0.5f*(G0a+G2a) + K96*(Fm0+Fm0p+Fj0+Fj0p);
            float bxy_j0 = 0.25f*(um0+um0p+uj0+uj0p);
            float Dp1 = 0.5f*(G1b+G3b) + K96*(Fj1+Fj1p+Fp1+Fp1p);
            float bxy_p1 = 0.25f*(uj1+uj1p+up1+up1p);
            float Ej1 = 0.5f*(G0b+G2b) - K96*(Fm1+Fm1p+Fj1+Fj1p);
            float bxy_j1 = 0.25f*(um1+um1p+uj1+uj1p);
            float Ep0 = 0.5f*(G1a+G3a) - K96*(Fj0+Fj0p+Fp0+Fp0p);
            float bxy_p0 = 0.25f*(uj0+uj0p+up0+up0p);
            float advec = 0.5f*( -(Bu1 - Bu0) - (Cu_jp - Cu_j)
                               + (Dp1*bxy_p1 - Dj0*bxy_j0)
                               + (Ep0*bxy_p0 - Ej1*bxy_j1) );
            float piw = (fDA*0.03125f)*( pim0*w[oT+jm*NXc+c0] + pim1*w[oT+jm*NXc+c1]
                                       + pip0*w[oT+jp*NXc+c0] + pip1*w[oT+jp*NXc+c1] );
            float trans = -piw * 0.25f*(u_zm + u_zp) * 32.0f;
            float sv    = sg[k]*(Pu[k]-Pp[k])*32.0f;
            float vbar  = 0.5f*(th[oT+j*NXc+c0] + th[oT+j*NXc+c1]) * sv;
            float dphix = phi[oT+j*NXc+c1] - phi[oT+j*NXc+c0];
            float press = fDY*( dphix*dpix*0.5f + dpix*(0.5f*fCPD)*vbar );
            float lap = u_zm + u_zp + um_i + up_i + u_iL + u_iR - 6.0f*u_c;
            out[rjg + ii] = (p0dA*u0[rjg + ii] + dt*(advec + trans + press + fDIFF*lap)) / p1dA;
            u_zm = u_c; u_c = u_zp;
        }
        async_fence();
        buf ^= 1;
    }
}

// ======================= v1 (x-tiled, async-LDS staged) ================
__device__ __forceinline__ void stage_v_tiles(const float* v, const float* u, int k,
        int r0, int r1, int R0, int R1, int R2, int R3, int x0, int t,
        float* sUb, float* sVb) {
    const int oU = k*SU, oV = k*SV;
    #pragma unroll
    for (int r = 0; r < 2; ++r) {                  // u rows r0,r1; 721-wrap
        const int row = r ? r1 : r0;
        const float* src = u + oU + row*(NXc+1);
        float* dst = sUb + r*LW;
        for (int c = t; c < 259; c += TB) {
            int col = x0 - 1 + c;
            if (col < 0) col += NXc+1; else if (col > NXc) col -= NXc+1;
            g2l(src + col, dst + c);
        }
    }
    #pragma unroll
    for (int r = 0; r < 4; ++r) {                  // v rows R0..R3; 720-wrap
        const int row = (r == 0) ? R0 : ((r == 1) ? R1 : ((r == 2) ? R2 : R3));
        const float* src = v + oV + row*NXc;
        float* dst = sVb + r*LW;
        for (int c = t; c < 258; c += TB) {
            int col = x0 - 1 + c;
            if (col < 0) col += NXc; else if (col >= NXc) col -= NXc;
            g2l(src + col, dst + c);
        }
    }
}

__global__ void v_kernel(const float* __restrict__ v,  const float* __restrict__ v0a,
                         const float* __restrict__ u,
                         const float* __restrict__ pi2d, const float* __restrict__ pi0,
                         const float* __restrict__ pi1v,
                         const float* __restrict__ w,   const float* __restrict__ phi,
                         const float* __restrict__ th,
                         const float* __restrict__ aF,  const float* __restrict__ bG,
                         const float* __restrict__ sg,  const float* __restrict__ Pu,
                         const float* __restrict__ Pp,
                         const float* __restrict__ dtp, float dtmul,
                         float* __restrict__ out) {
    __shared__ float shU[2*2*LW];
    __shared__ float shV[2*4*LW];
    const int t  = threadIdx.x;
    const int x0 = blockIdx.x * TB;
    const int j  = blockIdx.y;                     // 0..NY
    const int i  = x0 + t;
    const bool act = (i < NXc);
    const int ii = act ? i : (NXc - 1);
    const int r0 = wy(j-1), r1 = wy(j);
    const int R0 = r0, R1 = r0 + 1, R2 = r1, R3 = r1 + 1;
    const int jrow = (j == NYc) ? 1 : 2;           // staged V row matching row j
    const int cL = wx(ii-1), cR = wx(ii+1);
    const int qm = (ii == 0)       ? NXc : ii - 1; // 721-wrap
    const int qp = (ii == NXc - 1) ? 0   : ii + 2;
    const int jVm = (j == 0)   ? NYc : j - 1;      // 361-wrap
    const int jVp = (j == NYc) ? 0   : j + 1;
    const int a0 = r0*(NXc+1), a1 = r1*(NXc+1);
    float a0m = aF[a0+qm], a0i = aF[a0+ii], a0ip = aF[a0+ii+1], a0p = aF[a0+qp];
    float a1m = aF[a1+qm], a1i = aF[a1+ii], a1ip = aF[a1+ii+1], a1p = aF[a1+qp];
    float b0L = bG[R0*NXc+cL], b0C = bG[R0*NXc+ii], b0R = bG[R0*NXc+cR];
    float b1L = bG[R1*NXc+cL], b1C = bG[R1*NXc+ii], b1R = bG[R1*NXc+cR];
    float b2L = bG[R2*NXc+cL], b2C = bG[R2*NXc+ii], b2R = bG[R2*NXc+cR];
    float b3L = bG[R3*NXc+cL], b3C = bG[R3*NXc+ii], b3R = bG[R3*NXc+cR];
    float pi0L = pi2d[r0*NXc+cL], pi0R = pi2d[r0*NXc+cR];
    float pi1L = pi2d[r1*NXc+cL], pi1R = pi2d[r1*NXc+cR];
    float dpiy = pi2d[r1*NXc+ii] - pi2d[r0*NXc+ii];
    float p0dA = (fDA*0.03125f)*(pi0[r0*NXc+cL]+pi0[r0*NXc+cR]+pi0[r1*NXc+cL]+pi0[r1*NXc+cR]);
    float p1dA = (fDA*0.03125f)*(pi1v[r0*NXc+cL]+pi1v[r0*NXc+cR]+pi1v[r1*NXc+cL]+pi1v[r1*NXc+cR]);
    float dt = dtp[0] * dtmul;
    float v_zm = 0.0f;
    float v_c  = v[j*NXc + ii];
    stage_v_tiles(v, u, 0, r0, r1, R0, R1, R2, R3, x0, t, shU, shV);
    async_fence();
    int buf = 0;
    for (int k = 0; k < NZc; ++k) {
        if (k + 1 < NZc)
            stage_v_tiles(v, u, k+1, r0, r1, R0, R1, R2, R3, x0, t,
                          shU + (buf^1)*2*LW, shV + (buf^1)*4*LW);
        if (act) {
            const float* U0 = shU + buf*2*LW;      // u row r0
            const float* U1 = U0 + LW;             // u row r1
            const float* V0 = shV + buf*4*LW;      // v rows R0..R3
            const float* V1 = V0 + LW;
            const float* V2 = V1 + LW;
            const float* V3 = V2 + LW;
            const float* Vj = (jrow == 1) ? V1 : V2;
            const int oV = k*SV, oT = k*ST;
            const int vrj = oV + j*NXc;
            float v_zp = (k < NZc-1) ? v[vrj + SV + ii] : 0.0f;
            float F0m = a0m*U0[t], F0i = a0i*U0[t+1], F0ip = a0ip*U0[t+2], F0p = a0p*U0[t+3];
            float F1m = a1m*U1[t], F1i = a1i*U1[t+1], F1ip = a1ip*U1[t+2], F1p = a1p*U1[t+3];
            float v0L=V0[t], v0C=V0[t+1], v0R=V0[t+2];
            float v1L=V1[t], v1C=V1[t+1], v1R=V1[t+2];
            float v2L=V2[t], v2C=V2[t+1], v2R=V2[t+2];
            float v3L=V3[t], v3C=V3[t+1], v3R=V3[t+2];
            float vjL = Vj[t], vjR = Vj[t+2];
            float G0L=b0L*v0L, G0C=b0C*v0C, G0R=b0R*v0R;
            float G1L=b1L*v1L, G1C=b1C*v1C, G1R=b1R*v1R;
            float G2L=b2L*v2L, G2C=b2C*v2C, G2R=b2R*v2R;
            float G3L=b3L*v3L, G3C=b3C*v3C, G3R=b3R*v3R;
            float Q_i   = K48*(F0m + F0ip + F1m + F1ip);
            float Q_ip1 = K48*(F0i + F0p  + F1i + F1p);
            float termx = -( Q_ip1*(0.5f*(v_c + vjR)) - Q_i*(0.5f*(vjL + v_c)) );
            float R_a = K48*(G0L+G0R+G1L+G1R);
            float R_b = K48*(G2L+G2R+G3L+G3R);
            float termy = -( R_b*(0.5f*(v2C+v3C)) - R_a*(0.5f*(v0C+v1C)) );
            float S_r1_ip = 0.25f*(G2C+G2R+G3C+G3R) + K48*(F1i+F1p);
            float bxy_r1_ip = 0.25f*(v2C+v2R+v3C+v3R);
            float S_r0_i  = 0.25f*(G0L+G0C+G1L+G1C) + K48*(F0m+F0ip);
            float bxy_r0_i = 0.25f*(v0L+v0C+v1L+v1C);
            float T_r1_i  = 0.25f*(G2L+G2C+G3L+G3C) - K48*(F1m+F1ip);
            float bxy_r1_i = 0.25f*(v2L+v2C+v3L+v3C);
            float T_r0_ip = 0.25f*(G0C+G0R+G1C+G1R) - K48*(F0i+F0p);
            float bxy_r0_ip = 0.25f*(v0C+v0R+v1C+v1R);
            float termDE = (S_r1_ip*bxy_r1_ip - S_r0_i*bxy_r0_i)
                         + (T_r1_i*bxy_r1_i  - T_r0_ip*bxy_r0_ip);
            float advec = 0.5f*(termx + termy + termDE);
            float piw = (fDA*0.03125f)*( pi0L*w[oT+r0*NXc+cL] + pi0R*w[oT+r0*NXc+cR]
                                       + pi1L*w[oT+r1*NXc+cL] + pi1R*w[oT+r1*NXc+cR] );
            float trans = -piw * 0.25f*(v_zm + v_zp) * 32.0f;
            float sv    = sg[k]*(Pu[k]-Pp[k])*32.0f;
            float vbary = 0.5f*(th[oT+r0*NXc+ii] + th[oT+r1*NXc+ii]) * sv;
            float dphiy = phi[oT+r1*NXc+ii] - phi[oT+r0*NXc+ii];
            float press = fDX*( dphiy*dpiy*0.5f + dpiy*(0.5f*fCPD)*vbary );
            float lap = v_zm + v_zp + v[oV+jVm*NXc+ii] + v[oV+jVp*NXc+ii]
                      + vjL + vjR - 6.0f*v_c;
            out[vrj + ii] = (p0dA*v0a[vrj + ii] + dt*(advec + trans + press + fDIFF*lap)) / p1dA;
            v_zm = v_c; v_c = v_zp;
        }
        async_fence();
        buf ^= 1;
    }
}

// ================================================================= launch
extern "C" void kernel_launch(void* const* d_in, const int* in_sizes, int n_in,
                              void* d_out, int out_size, void* d_ws, size_t ws_size,
                              hipStream_t stream) {
    const float* pi  = (const float*)d_in[0];
    const float* th  = (const float*)d_in[1];
    const float* u   = (const float*)d_in[2];
    const float* v   = (const float*)d_in[3];
    const float* w   = (const float*)d_in[4];
    const float* phi = (const float*)d_in[5];
    const float* Y   = (const float*)d_in[6];
    const float* sg  = (const float*)d_in[8];
    const float* Pu  = (const float*)d_in[9];
    const float* Pp  = (const float*)d_in[10];
    const float* dtp = (const float*)d_in[12];
    (void)in_sizes; (void)n_in; (void)out_size; (void)ws_size;

    const int SZ_PI = NYc*NXc;
    const int SZ_TH = NZc*NYc*NXc;
    const int SZ_U  = NZc*NYc*(NXc+1);
    const int SZ_V  = NZc*(NYc+1)*NXc;

    float* out  = (float*)d_out;
    float* pi_o = out;
    float* th_o = out + SZ_PI;
    float* u_o  = th_o + SZ_TH;
    float* v_o  = u_o + SZ_U;

    float* ws   = (float*)d_ws;                 // needs ~103 MB
    float* pi_1 = ws;
    float* th_1 = pi_1 + SZ_PI;
    float* u_1  = th_1 + SZ_TH;
    float* v_1  = u_1 + SZ_U;
    float* cosY = v_1 + SZ_V;
    float* aF   = cosY + 384;
    float* bG   = aF + NYc*(NXc+1);

    dim3 blk(TB);
    int gC  = ((NYc+1)*NXc + TB-1)/TB;
    int gPI = (SZ_PI + TB-1)/TB;
    dim3 gTH(3, NYc);        // 3 x-tiles of 256 cover 720
    dim3 gUU(3, NYc);        // 3 x-tiles cover 721
    dim3 gVV(3, NYc+1);

    cosy_kernel<<<2, blk, 0, stream>>>(Y, cosY);

    // ---- step 1: dt/2, current == "0" state == inputs -> intermediates in ws
    coeff_kernel<<<gC, blk, 0, stream>>>(pi, cosY, aF, bG);
    pi_kernel   <<<gPI, blk, 0, stream>>>(pi, u, v, aF, bG, dtp, 0.5f, pi_1);
    theta_kernel<<<gTH, blk, 0, stream>>>(th, th, pi, pi, pi_1, u, v, w, aF, bG,
                                          dtp, 0.5f, th_1);
    u_kernel    <<<gUU, blk, 0, stream>>>(u, u, v, pi, pi, pi_1, w, phi, th,
                                          aF, bG, sg, Pu, Pp, dtp, 0.5f, u_1);
    v_kernel    <<<gVV, blk, 0, stream>>>(v, v, u, pi, pi, pi_1, w, phi, th,
                                          aF, bG, sg, Pu, Pp, dtp, 0.5f, v_1);

    // ---- step 2: dt, current = intermediates, "0" state = inputs -> d_out
    coeff_kernel<<<gC, blk, 0, stream>>>(pi_1, cosY, aF, bG);
    pi_kernel   <<<gPI, blk, 0, stream>>>(pi, u_1, v_1, aF, bG, dtp, 1.0f, pi_o);
    theta_kernel<<<gTH, blk, 0, stream>>>(th_1, th, pi_1, pi, pi_o, u_1, v_1, w,
                                          aF, bG, dtp, 1.0f, th_o);
    u_kernel    <<<gUU, blk, 0, stream>>>(u_1, u, v_1, pi_1, pi, pi_o, w, phi, th_1,
                                          aF, bG, sg, Pu, Pp, dtp, 1.0f, u_o);
    v_kernel    <<<gVV, blk, 0, stream>>>(v_1, v, u_1, pi_1, pi, pi_o, w, phi, th_1,
                                          aF, bG, sg, Pu, Pp, dtp, 1.0f, v_o);
}